// Mass_conservation_loss_87290915324264
// MI455X (gfx1250) — compile-verified
//
#include <hip/hip_runtime.h>

typedef __attribute__((ext_vector_type(2))) float v2f;
typedef __attribute__((ext_vector_type(8))) float v8f;

// ---------------------------------------------------------------------------
// Kernel 1: zero the node accumulator (d_ws) and the scalar output.
// num_nodes arrives as a device-side scalar (in_sizes[2]==1), so the bound is
// read inside the kernel and we use a grid-stride loop with a fixed grid.
// ---------------------------------------------------------------------------
__global__ void mc_zero(float* __restrict__ acc,
                        const int* __restrict__ n_ptr,
                        float* __restrict__ out) {
  const int N = *n_ptr;
  const int stride = (int)(gridDim.x * blockDim.x);
  for (int i = (int)(blockIdx.x * blockDim.x + threadIdx.x); i < N; i += stride)
    acc[i] = 0.0f;
  if (blockIdx.x == 0 && threadIdx.x == 0) out[0] = 0.0f;
}

// ---------------------------------------------------------------------------
// Kernel 2: scatter. Each thread handles 2 edges: one coalesced int4 load
// (src0,dst0,src1,dst1) + one float2 load, then 4 non-returning fp32 atomics
// (native global_atomic_add_f32, STOREcnt path). The 4 MB accumulator is
// L2-resident (192 MB L2), so the 2E = 40M atomics never touch HBM.
// HBM traffic = 240 MB stream -> ~10.3 us floor at 23.3 TB/s.
// ---------------------------------------------------------------------------
__global__ void mc_scatter(const int* __restrict__ edges,
                           const float* __restrict__ values,
                           float* __restrict__ acc,
                           const int E) {
  const int P = E >> 1;  // edge pairs
  const int i = (int)(blockIdx.x * blockDim.x + threadIdx.x);
  if (i < P) {
    const int4   e = ((const int4*)edges)[i];
    const float2 v = ((const float2*)values)[i];
    unsafeAtomicAdd(&acc[e.x],  v.x);
    unsafeAtomicAdd(&acc[e.y], -v.x);
    unsafeAtomicAdd(&acc[e.z],  v.y);
    unsafeAtomicAdd(&acc[e.w], -v.y);
  }
  if (i == 0 && (E & 1)) {  // odd tail edge, if any
    const int   s = edges[2 * (E - 1)];
    const int   d = edges[2 * (E - 1) + 1];
    const float v = values[E - 1];
    unsafeAtomicAdd(&acc[s],  v);
    unsafeAtomicAdd(&acc[d], -v);
  }
}

// ---------------------------------------------------------------------------
// Kernel 3: reduce acc[0..N) -> out[0] using V_WMMA_F32_16X16X4_F32.
// B = all-ones => D[m][n] = sum_k A[m][k] + C[m][n]. Each wave folds 64 floats
// per WMMA into a persistent 16x16 f32 accumulator. Hot loop is exactly one
// coalesced global_load_b64 + one v_wmma per chunk; all loop branches are
// wave-uniform so EXEC is all-1s at every WMMA (ISA requirement).
// Each row-sum is replicated across 16 columns, so the cross-lane total is
// 16x the true sum -> scale by 1/16 (exact). The <=63-element tail is summed
// scalar by wave 0 and folded in with an exact x16 weight.
// ---------------------------------------------------------------------------
__global__ void mc_reduce(const float* __restrict__ acc,
                          const int* __restrict__ n_ptr,
                          float* __restrict__ out) {
  const int N = *n_ptr;
  const int lane   = (int)(threadIdx.x & 31u);
  const int wave   = (int)((blockIdx.x * blockDim.x + threadIdx.x) >> 5);
  const int nwaves = (int)((gridDim.x * blockDim.x) >> 5);
  const int full   = N >> 6;  // full 64-float chunks (no bounds checks inside)

  v8f c = {};                 // 16x16 f32 accumulator (8 VGPRs)
  v2f b;                      // 4x16 ones matrix (2 VGPRs)
  b.x = 1.0f; b.y = 1.0f;

  for (int ch = wave; ch < full; ch += nwaves) {
    const v2f a = *(const v2f*)(acc + (ch << 6) + (lane << 1));  // b64 load
    // 8 args: (neg_a, A, neg_b, B, c_mod, C, reuse_a, reuse_b)
    c = __builtin_amdgcn_wmma_f32_16x16x4_f32(
        /*neg_a=*/false, a, /*neg_b=*/false, b,
        /*c_mod=*/(short)0, c, /*reuse_a=*/false, /*reuse_b=*/false);
  }

  float s = c[0] + c[1] + c[2] + c[3] + c[4] + c[5] + c[6] + c[7];

  if (wave == 0) {            // scalar tail: elements [full*64, N)
    float t = 0.0f;
    for (int i = (full << 6) + lane; i < N; i += 32) t += acc[i];
    s += t * 16.0f;           // exact reweight to match WMMA's 16x replication
  }

  // wave32 cross-lane reduction
  for (int off = 16; off > 0; off >>= 1)
    s += __shfl_down(s, off, 32);
  if (lane == 0)
    unsafeAtomicAdd(out, s * 0.0625f);  // /16: column replication factor
}

// ---------------------------------------------------------------------------
// Host-side launcher. d_in[0]=edges (int32, 2E), d_in[1]=values (f32, E),
// d_in[2]=num_nodes (int32 scalar on device). d_out = 1 float. d_ws holds the
// N-float accumulator (4 MB for N=1e6).
// ---------------------------------------------------------------------------
extern "C" void kernel_launch(void* const* d_in, const int* in_sizes, int n_in,
                              void* d_out, int out_size, void* d_ws, size_t ws_size,
                              hipStream_t stream) {
  const int*   edges  = (const int*)d_in[0];
  const float* values = (const float*)d_in[1];
  const int*   n_ptr  = (const int*)d_in[2];
  float*       out    = (float*)d_out;
  float*       acc    = (float*)d_ws;

  const int E = in_sizes[1];      // number of edges (values is length E)
  const int P = E >> 1;           // edge pairs, one per scatter thread

  mc_zero<<<2048, 256, 0, stream>>>(acc, n_ptr, out);
  mc_scatter<<<(P + 255) / 256, 256, 0, stream>>>(edges, values, acc, E);
  mc_reduce<<<1024, 256, 0, stream>>>(acc, n_ptr, out);
}